// net_gram_68307159875995
// MI455X (gfx1250) — compile-verified
//
#include <hip/hip_runtime.h>
#include <hip/hip_bf16.h>

// ---------------------------------------------------------------------------
// Sizes (fixed by the reference)
// ---------------------------------------------------------------------------
#define B_  4
#define C_  256
#define T_  64
#define F_  512
#define H_  8
#define TF_ (T_ * F_)          // 32768 floats per (b,c) plane
#define BC_ (B_ * C_)          // 1024

typedef __attribute__((ext_vector_type(2))) float v2f;
typedef __attribute__((ext_vector_type(4))) float v4f;
typedef __attribute__((ext_vector_type(8))) float v8f;
typedef __attribute__((ext_vector_type(4))) int   v4i;

typedef __attribute__((address_space(1))) v4i gv4i;   // global b128 payload
typedef __attribute__((address_space(3))) v4i lv4i;   // LDS b128 payload

// V_WMMA_F32_16X16X4_F32 : D(16x16 f32) = A(16x4) * B(4x16) + C
__device__ __forceinline__ v8f wmma4(v2f a, v2f b, v8f c) {
    return __builtin_amdgcn_wmma_f32_16x16x4_f32(false, a, false, b,
                                                 (short)0, c, false, false);
}

__device__ __forceinline__ v2f ldf2(const float* p) { return *(const v2f*)p; }

// ---- gfx1250 async global->LDS path (guarded) -----------------------------
#if __has_builtin(__builtin_amdgcn_global_load_async_to_lds_b128)
#define HAVE_ASYNC_LDS 1
#else
#define HAVE_ASYNC_LDS 0
#endif

#if __has_builtin(__builtin_amdgcn_s_wait_asynccnt)
#define WAIT_ASYNCCNT(n) __builtin_amdgcn_s_wait_asynccnt(n)
#else
#define WAIT_ASYNCCNT(n) asm volatile("s_wait_asynccnt %0" :: "i"(n) : "memory")
#endif

// ---------------------------------------------------------------------------
// K1: xm[b,c,f] = mean over t of x[b,c,t,f]
// ---------------------------------------------------------------------------
__global__ void k1_mean_t(const float* __restrict__ x, float* __restrict__ xm) {
    int idx = blockIdx.x * blockDim.x + threadIdx.x;      // < BC_*F_
    int f  = idx & (F_ - 1);
    int bc = idx >> 9;                                    // F_ = 512
    const float* p = x + (size_t)bc * TF_ + f;
    float s = 0.f;
#pragma unroll 8
    for (int t = 0; t < T_; ++t) s += p[t * F_];
    xm[idx] = s * (1.0f / (float)T_);
}

// ---------------------------------------------------------------------------
// K2: qs = (xm @ Wq^T + bq) * 0.125 ;  kb = xm @ Wk^T       (WMMA f32)
// ---------------------------------------------------------------------------
__global__ void k2_qk(const float* __restrict__ xm,
                      const float* __restrict__ Wq, const float* __restrict__ bq,
                      const float* __restrict__ Wk,
                      float* __restrict__ qs, float* __restrict__ kb) {
    int wid   = blockIdx.x * 4 + (threadIdx.x >> 5);      // 0..2047
    int mtile = wid >> 5;                                 // 64 M tiles
    int ntile = wid & 31;                                 // 32 N tiles
    int m0 = mtile * 16, n0 = ntile * 16;
    int lane = threadIdx.x & 31;
    int hi = lane >> 4, ln = lane & 15;

    v8f aq = {}, ak = {};
    int arow = (m0 + ln) * F_ + 2 * hi;
    int brow = (n0 + ln) * F_ + 2 * hi;
#pragma unroll 4
    for (int k0 = 0; k0 < F_; k0 += 4) {
        v2f a   = ldf2(&xm[arow + k0]);
        v2f bqv = ldf2(&Wq[brow + k0]);
        v2f bkv = ldf2(&Wk[brow + k0]);
        aq = wmma4(a, bqv, aq);
        ak = wmma4(a, bkv, ak);
    }
    int col  = n0 + ln;
    float bias = bq[col];
#pragma unroll
    for (int r = 0; r < 8; ++r) {
        int row = m0 + r + 8 * hi;
        qs[row * F_ + col] = (aq[r] + bias) * 0.125f;     // fold scale^2 = 1/8
        kb[row * F_ + col] = ak[r];
    }
}

// ---------------------------------------------------------------------------
// K3: per (b,g): logits = qs_g @ k_g^T (256x256x64) + row softmax -> w
// ---------------------------------------------------------------------------
__global__ void k3_attn(const float* __restrict__ qs, const float* __restrict__ kb,
                        float* __restrict__ wbuf) {
    int wid   = blockIdx.x * 4 + (threadIdx.x >> 5);      // 0..511
    int bg    = wid >> 4;
    int strip = wid & 15;
    int b = bg >> 3, g = bg & 7;
    int m0 = strip * 16;
    int lane = threadIdx.x & 31;
    int hi = lane >> 4, ln = lane & 15;

    const float* qb = qs + (size_t)(b * C_) * F_ + g * 64;
    const float* kB = kb + (size_t)(b * C_) * F_ + g * 64;

    v8f acc[16];
#pragma unroll
    for (int j = 0; j < 16; ++j) acc[j] = (v8f){};

#pragma unroll
    for (int k0 = 0; k0 < 64; k0 += 4) {
        v2f a = ldf2(&qb[(m0 + ln) * F_ + k0 + 2 * hi]);
#pragma unroll
        for (int j = 0; j < 16; ++j) {
            v2f bf = ldf2(&kB[(j * 16 + ln) * F_ + k0 + 2 * hi]); // k^T trick
            acc[j] = wmma4(a, bf, acc[j]);
        }
    }

    float* wrow = wbuf + (size_t)bg * C_ * C_;
#pragma unroll
    for (int r = 0; r < 8; ++r) {
        float mx = -3.402823466e+38f;
#pragma unroll
        for (int j = 0; j < 16; ++j) mx = fmaxf(mx, acc[j][r]);
#pragma unroll
        for (int m = 1; m < 16; m <<= 1) mx = fmaxf(mx, __shfl_xor(mx, m, 32));
        float s = 0.f;
#pragma unroll
        for (int j = 0; j < 16; ++j) {
            float e = __expf(acc[j][r] - mx);
            acc[j][r] = e;
            s += e;
        }
#pragma unroll
        for (int m = 1; m < 16; m <<= 1) s += __shfl_xor(s, m, 32);
        float inv = 1.0f / s;
        int row = m0 + r + 8 * hi;
#pragma unroll
        for (int j = 0; j < 16; ++j)
            wrow[row * C_ + j * 16 + ln] = acc[j][r] * inv;
    }
}

// ---------------------------------------------------------------------------
// K4a: mg[b,e,g] = (1/F) * sum_{f in head g} xm[b,e,f]
// ---------------------------------------------------------------------------
__global__ void k4a_mg(const float* __restrict__ xm, float* __restrict__ mg) {
    int idx = blockIdx.x * blockDim.x + threadIdx.x;      // < B_*C_*H_
    int g = idx & 7, be = idx >> 3;
    const float* p = xm + (size_t)be * F_ + g * 64;
    float s = 0.f;
#pragma unroll 8
    for (int fl = 0; fl < 64; ++fl) s += p[fl];
    mg[idx] = s * (1.0f / (float)F_);
}

// ---------------------------------------------------------------------------
// K4b: exact SE factor, computed BEFORE the big GEMM
// ---------------------------------------------------------------------------
__global__ void k4b_se(const float* __restrict__ wbuf, const float* __restrict__ mg,
                       const float* __restrict__ W1, const float* __restrict__ W2,
                       float* __restrict__ sfac) {
    __shared__ float s0[C_];
    __shared__ float s1[16];
    int b = blockIdx.x, c = threadIdx.x;

    float acc = 0.f;
    for (int g = 0; g < H_; ++g) {
        const float* wr  = wbuf + ((size_t)(b * H_ + g) * C_ + c) * C_;
        const float* mgb = mg + (size_t)b * C_ * H_ + g;
#pragma unroll 4
        for (int e = 0; e < C_; ++e) acc += wr[e] * mgb[e * H_];
    }
    s0[c] = acc;
    __syncthreads();
    if (c < 16) {
        float a2 = 0.f;
        for (int e = 0; e < C_; ++e) a2 += W1[c * C_ + e] * s0[e];
        s1[c] = fmaxf(a2, 0.f);
    }
    __syncthreads();
    float z = 0.f;
#pragma unroll
    for (int r = 0; r < 16; ++r) z += W2[c * 16 + r] * s1[r];
    sfac[b * C_ + c] = 1.0f / (1.0f + __expf(-z));
}

// ---------------------------------------------------------------------------
// K5: fused attention-apply + SE scale (dominant GEMM, 17.2 GFLOP f32)
//   out[b,c,t,f] = sfac[b,c] * sum_e w[b, f/64, c, e] * x[b,e,t,f]
//
// Block = 128 thr = 4 waves, block tile 128 rows x 64 cols (one t).
// All 4 waves share the same X columns -> X K-slabs (32x64) are staged in LDS
// via GLOBAL_LOAD_ASYNC_TO_LDS_B128 with double buffering (ASYNCcnt), padded
// to stride 72 so B-fragment ds reads are bank-conflict free.
// ---------------------------------------------------------------------------
#define KB_   32                  // K-slab depth
#define LSTR  72                  // padded LDS row stride (floats)
#define BUFSZ (KB_ * LSTR)        // floats per buffer

__global__ void k5_apply(const float* __restrict__ x, const float* __restrict__ wbuf,
                         const float* __restrict__ sfac, float* __restrict__ out) {
    __shared__ float sx[2][BUFSZ];           // 2 x 9 KB

    int nt = blockIdx.x;                 // t
    int mt = blockIdx.y;                 // 128-row strip of c
    int bg = blockIdx.z;
    int b = bg >> 3, g = bg & 7;
    int wave = threadIdx.x >> 5;
    int lane = threadIdx.x & 31;
    int hi = lane >> 4, ln = lane & 15;
    int mbase = mt * 128 + wave * 32;    // wave's first output row (c)

    const float* wrow = wbuf + (size_t)bg * C_ * C_;
    const float* xb = x   + (size_t)b * C_ * TF_ + (size_t)nt * F_ + g * 64;
    float*       ob = out + (size_t)b * C_ * TF_ + (size_t)nt * F_ + g * 64;

    // per-thread staging coordinates: row r (0..31), 16-col quarter q (0..3)
    int sr = threadIdx.x >> 2, sq = threadIdx.x & 3;
    const float* gp0 = xb + (size_t)sr * TF_ + sq * 16;

    v8f acc[2][4];
#pragma unroll
    for (int mi = 0; mi < 2; ++mi)
#pragma unroll
        for (int ni = 0; ni < 4; ++ni) acc[mi][ni] = (v8f){};

    int ar0 = (mbase + ln) * C_ + 2 * hi;
    int ar1 = (mbase + 16 + ln) * C_ + 2 * hi;

    // ---- slab issue helper (4 x b128 per thread, imm offsets 0/16/32/48 B)
#if HAVE_ASYNC_LDS
#define ISSUE_SLAB(kb, buf)                                                        \
    do {                                                                           \
        gv4i* gp = (gv4i*)(gp0 + (size_t)(kb) * KB_ * TF_);                        \
        lv4i* lp = (lv4i*)&sx[(buf)][sr * LSTR + sq * 16];                         \
        __builtin_amdgcn_global_load_async_to_lds_b128(gp, lp, 0, 0);              \
        __builtin_amdgcn_global_load_async_to_lds_b128(gp, lp, 16, 0);             \
        __builtin_amdgcn_global_load_async_to_lds_b128(gp, lp, 32, 0);             \
        __builtin_amdgcn_global_load_async_to_lds_b128(gp, lp, 48, 0);             \
    } while (0)
#else
#define ISSUE_SLAB(kb, buf)                                                        \
    do {                                                                           \
        const float* gp = gp0 + (size_t)(kb) * KB_ * TF_;                          \
        float* lp = &sx[(buf)][sr * LSTR + sq * 16];                               \
        v4f t0 = *(const v4f*)(gp + 0);                                            \
        v4f t1 = *(const v4f*)(gp + 4);                                            \
        v4f t2 = *(const v4f*)(gp + 8);                                            \
        v4f t3 = *(const v4f*)(gp + 12);                                           \
        *(v4f*)(lp + 0) = t0; *(v4f*)(lp + 4) = t1;                                \
        *(v4f*)(lp + 8) = t2; *(v4f*)(lp + 12) = t3;                               \
    } while (0)
#endif

    ISSUE_SLAB(0, 0);                                    // prologue

    const int NKB = C_ / KB_;                            // 8 slabs
    for (int kbi = 0; kbi < NKB; ++kbi) {
        int buf = kbi & 1;
        if (kbi + 1 < NKB) {
            ISSUE_SLAB(kbi + 1, buf ^ 1);
#if HAVE_ASYNC_LDS
            WAIT_ASYNCCNT(4);                            // current slab landed
#endif
        } else {
#if HAVE_ASYNC_LDS
            WAIT_ASYNCCNT(0);
#endif
        }
        __syncthreads();                                 // slab visible to all

        const float* sb = sx[buf];
        int kg = kbi * KB_;
#pragma unroll
        for (int kk = 0; kk < KB_; kk += 4) {
            v2f a0 = ldf2(&wrow[ar0 + kg + kk]);
            v2f a1 = ldf2(&wrow[ar1 + kg + kk]);
            int rbase = (kk + 2 * hi) * LSTR + ln;
#pragma unroll
            for (int ni = 0; ni < 4; ++ni) {
                v2f bf;
                bf.x = sb[rbase + ni * 16];              // B[kk+2hi  , n]
                bf.y = sb[rbase + LSTR + ni * 16];       // B[kk+2hi+1, n]
                acc[0][ni] = wmma4(a0, bf, acc[0][ni]);
                acc[1][ni] = wmma4(a1, bf, acc[1][ni]);
            }
        }
        __syncthreads();                                 // done reading buf
    }

    // epilogue: SE scale + store (contiguous across each 16-lane group)
#pragma unroll
    for (int mi = 0; mi < 2; ++mi) {
#pragma unroll
        for (int r = 0; r < 8; ++r) {
            int row = mbase + mi * 16 + r + 8 * hi;
            float sf = sfac[b * C_ + row];
            float* orow = ob + (size_t)row * TF_ + ln;
#pragma unroll
            for (int ni = 0; ni < 4; ++ni)
                orow[ni * 16] = acc[mi][ni][r] * sf;
        }
    }
#undef ISSUE_SLAB
}

// ---------------------------------------------------------------------------
// Launch
// ---------------------------------------------------------------------------
extern "C" void kernel_launch(void* const* d_in, const int* in_sizes, int n_in,
                              void* d_out, int out_size, void* d_ws, size_t ws_size,
                              hipStream_t stream) {
    const float* x  = (const float*)d_in[0];
    const float* Wq = (const float*)d_in[1];
    const float* bq = (const float*)d_in[2];
    const float* Wk = (const float*)d_in[3];
    const float* W1 = (const float*)d_in[4];
    const float* W2 = (const float*)d_in[5];
    float* out = (float*)d_out;

    // workspace layout (floats): xm | qs | kb | w | mg | sfac  (~14.1 MB)
    float* ws   = (float*)d_ws;
    float* xm   = ws;                                  // BC_*F_   = 524288
    float* qs   = xm + (size_t)BC_ * F_;               // 524288
    float* kbuf = qs + (size_t)BC_ * F_;               // 524288
    float* wbuf = kbuf + (size_t)BC_ * F_;             // B*H*C*C  = 2097152
    float* mg   = wbuf + (size_t)B_ * H_ * C_ * C_;    // B*C*H    = 8192
    float* sfac = mg + (size_t)B_ * C_ * H_;           // B*C      = 1024

    k1_mean_t<<<(BC_ * F_) / 256, 256, 0, stream>>>(x, xm);
    k2_qk    <<<512, 128, 0, stream>>>(xm, Wq, bq, Wk, qs, kbuf);
    k3_attn  <<<128, 128, 0, stream>>>(qs, kbuf, wbuf);
    k4a_mg   <<<(B_ * C_ * H_) / 256, 256, 0, stream>>>(xm, mg);
    k4b_se   <<<B_, C_, 0, stream>>>(wbuf, mg, W1, W2, sfac);
    k5_apply <<<dim3(T_, 2, B_ * H_), 128, 0, stream>>>(x, wbuf, sfac, out);
}